// ESN_Cell_13202729468549
// MI455X (gfx1250) — compile-verified
//
#include <hip/hip_runtime.h>
#include <cstdint>
#include <cstddef>

// Problem constants (match reference)
constexpr int   kB     = 512;
constexpr int   kD     = 256;
constexpr int   kN     = 8192;
constexpr int   kNNZ   = 134217;
constexpr int   kNNZ4  = kNNZ & ~3;   // 134216, vectorized portion
constexpr float kAlpha = 0.9f;

constexpr int kG  = 4;     // batch rows per workgroup
constexpr int kNT = 1024;  // threads per workgroup (32 wave32s)

// Dynamic LDS layout: s_state[kG*kN] | s_acc[kG*kN] | s_in[kG*kD]
// = 128KB + 128KB + 4KB = 260KB  (<= 320KB WGP LDS)

__device__ __forceinline__ float esn_tanh(float x) {
#if __has_builtin(__builtin_amdgcn_tanhf)
    return __builtin_amdgcn_tanhf(x);   // v_tanh_f32 (CDNA5 native transcendental)
#else
    return tanhf(x);
#endif
}

__global__ __launch_bounds__(kNT, 1)
void esn_fused_kernel(const float* __restrict__ inputs,     // [B,D]
                      const float* __restrict__ state,      // [B,N]
                      const float* __restrict__ win_vals,   // [N]
                      const float* __restrict__ win_bias,   // [N]
                      const float* __restrict__ wres_vals,  // [NNZ]
                      const int*   __restrict__ win_rows,   // [N]
                      const int*   __restrict__ wres_rows,  // [NNZ]
                      const int*   __restrict__ wres_cols,  // [NNZ]
                      float* __restrict__ out)               // [B,N]
{
    extern __shared__ float smem[];
    float* s_state = smem;                 // kG * kN
    float* s_acc   = smem + kG * kN;       // kG * kN
    float* s_in    = smem + 2 * kG * kN;   // kG * kD

    const int tid = threadIdx.x;
    const int b0  = blockIdx.x * kG;

    // ---- Phase 1: async-copy G contiguous state rows (128 KB) and G input
    //      rows (4 KB) into LDS, 16 bytes per lane per issue. ASYNCcnt tracks
    //      completion; each wave waits its own counter, barrier publishes.
    {
        const float* gsrc = state + (size_t)b0 * kN;
        const int nvec = (kG * kN) / 4;            // 8192 x 16B chunks
        for (int i = tid; i < nvec; i += kNT) {
            unsigned lds_off = (unsigned)(uintptr_t)(s_state + i * 4);
            const float* ga  = gsrc + i * 4;
            asm volatile("global_load_async_to_lds_b128 %0, %1, off"
                         :: "v"(lds_off), "v"(ga) : "memory");
        }
        const float* gin = inputs + (size_t)b0 * kD;
        const int nvin = (kG * kD) / 4;            // 256 x 16B chunks
        if (tid < nvin) {
            unsigned lds_off = (unsigned)(uintptr_t)(s_in + tid * 4);
            const float* ga  = gin + tid * 4;
            asm volatile("global_load_async_to_lds_b128 %0, %1, off"
                         :: "v"(lds_off), "v"(ga) : "memory");
        }
        asm volatile("s_wait_asynccnt 0x0" ::: "memory");
    }
    __syncthreads();

    // ---- Phase 2: init accumulators with the sparse input projection:
    //      acc[g][j] = inputs[b0+g, win_rows[j]] * win_vals[j] + win_bias[j]
    //      Vectorized: b128 loads of the triplet streams, b128 LDS stores.
    for (int j = tid * 4; j < kN; j += kNT * 4) {
        const int4   rw = *(const int4*)  (win_rows + j);
        const float4 wv = *(const float4*)(win_vals + j);
        const float4 wb = *(const float4*)(win_bias + j);
#pragma unroll
        for (int g = 0; g < kG; ++g) {
            const float* in_g = s_in + g * kD;
            float4 a;
            a.x = fmaf(in_g[rw.x], wv.x, wb.x);
            a.y = fmaf(in_g[rw.y], wv.y, wb.y);
            a.z = fmaf(in_g[rw.z], wv.z, wb.z);
            a.w = fmaf(in_g[rw.w], wv.w, wb.w);
            *(float4*)(s_acc + g * kN + j) = a;    // ds_store_b128
        }
    }
    __syncthreads();

    // ---- Phase 3: COO SpMM, entirely inside LDS.
    //      Triplet stream loaded as b128 (4 nnz per trip); 16 independent
    //      ds_load_b32 + 16 ds_add_f32 per trip to hide LDS latency.
    for (int k = tid * 4; k < kNNZ4; k += kNT * 4) {
        const int4   r4 = *(const int4*)  (wres_rows + k);
        const int4   c4 = *(const int4*)  (wres_cols + k);
        const float4 v4 = *(const float4*)(wres_vals + k);
#pragma unroll
        for (int g = 0; g < kG; ++g) {
            const float* st_g = s_state + g * kN;
            float*       ac_g = s_acc   + g * kN;
            atomicAdd(&ac_g[c4.x], st_g[r4.x] * v4.x);   // -> ds_add_f32
            atomicAdd(&ac_g[c4.y], st_g[r4.y] * v4.y);
            atomicAdd(&ac_g[c4.z], st_g[r4.z] * v4.z);
            atomicAdd(&ac_g[c4.w], st_g[r4.w] * v4.w);
        }
    }
    // tail (kNNZ % 4 == 1 element)
    for (int k = kNNZ4 + tid; k < kNNZ; k += kNT) {
        const int   r = wres_rows[k];
        const int   c = wres_cols[k];
        const float v = wres_vals[k];
#pragma unroll
        for (int g = 0; g < kG; ++g)
            atomicAdd(&s_acc[g * kN + c], s_state[g * kN + r] * v);
    }
    __syncthreads();

    // ---- Phase 4: tanh + leaky integration, b128 LDS reads + b128 writeback.
    for (int j = tid * 4; j < kN; j += kNT * 4) {
#pragma unroll
        for (int g = 0; g < kG; ++g) {
            const float4 st = *(const float4*)(s_state + g * kN + j);
            const float4 ac = *(const float4*)(s_acc   + g * kN + j);
            float4 r;
            r.x = fmaf(kAlpha, esn_tanh(ac.x) - st.x, st.x);
            r.y = fmaf(kAlpha, esn_tanh(ac.y) - st.y, st.y);
            r.z = fmaf(kAlpha, esn_tanh(ac.z) - st.z, st.z);
            r.w = fmaf(kAlpha, esn_tanh(ac.w) - st.w, st.w);
            *(float4*)(out + (size_t)(b0 + g) * kN + j) = r;  // global_store_b128
        }
    }
}

extern "C" void kernel_launch(void* const* d_in, const int* in_sizes, int n_in,
                              void* d_out, int out_size, void* d_ws, size_t ws_size,
                              hipStream_t stream) {
    (void)in_sizes; (void)n_in; (void)d_ws; (void)ws_size; (void)out_size;

    const float* inputs    = (const float*)d_in[0];
    const float* state     = (const float*)d_in[1];
    const float* win_vals  = (const float*)d_in[2];
    const float* win_bias  = (const float*)d_in[3];
    const float* wres_vals = (const float*)d_in[4];
    const int*   win_rows  = (const int*)d_in[5];
    const int*   wres_rows = (const int*)d_in[6];
    const int*   wres_cols = (const int*)d_in[7];
    float*       out       = (float*)d_out;

    const size_t lds_bytes = (size_t)(2 * kG * kN + kG * kD) * sizeof(float); // 266,240 B
    hipFuncSetAttribute((const void*)esn_fused_kernel,
                        hipFuncAttributeMaxDynamicSharedMemorySize,
                        (int)lds_bytes);

    dim3 grid(kB / kG);   // 128 workgroups, one per WGP
    dim3 block(kNT);      // 1024 threads = 32 wave32s
    esn_fused_kernel<<<grid, block, lds_bytes, stream>>>(
        inputs, state, win_vals, win_bias, wres_vals,
        win_rows, wres_rows, wres_cols, out);
}